// DifferentiableICP_70798240907884
// MI455X (gfx1250) — compile-verified
//
#include <hip/hip_runtime.h>
#include <hip/hip_bf16.h>
#include <math.h>

typedef __attribute__((ext_vector_type(2))) float v2f;
typedef __attribute__((ext_vector_type(8))) float v8f;

#define BATCH 32
#define NPTS 2048
#define MPTS 2048
#define ITERS 5
#define ROWS_PER_BLOCK 128
#define WAVES 8
#define NBLK (NPTS / ROWS_PER_BLOCK)   // 16 row-tiles per batch

// ---------------------------------------------------------------------------
// Hot kernel: for a 128-row tile of st = R*src + t, compute softmax
// correspondences against all M targets via WMMA-generated logits, produce
// target_corr rows, and emit deterministic per-block partial sums
// (sum st, sum tc, sum st (x) tc) for the Kabsch update.
// Logit trick: softmax(-dist) drops s_sq, logit = 2*cross - tsq, computed as
//   A row = [2*stx, 2*sty, 1, 0],  B col = [tx, ty, -tsq, 0]  (K=4 f32 WMMA)
// ---------------------------------------------------------------------------
__global__ __launch_bounds__(256) void icp_corr_kernel(
    const float* __restrict__ src, const float* __restrict__ tgt,
    const float* __restrict__ T, float* __restrict__ partial) {
  __shared__ float s_tx[MPTS], s_ty[MPTS], s_tsq[MPTS];
  __shared__ float s_stx[ROWS_PER_BLOCK], s_sty[ROWS_PER_BLOCK];
  __shared__ float s_tcx[ROWS_PER_BLOCK], s_tcy[ROWS_PER_BLOCK];
  __shared__ float s_red[8][WAVES];

  const int b    = blockIdx.y;
  const int tid  = threadIdx.x;
  const int wave = tid >> 5;
  const int lane = tid & 31;
  const int l15  = lane & 15;
  const bool hi  = lane >= 16;

  // Stage target batch slice into LDS (tx, ty, |t|^2)
  const float* tb = tgt + (size_t)b * MPTS * 2;
  for (int i = tid; i < MPTS; i += 256) {
    float x = tb[2 * i], y = tb[2 * i + 1];
    s_tx[i] = x; s_ty[i] = y; s_tsq[i] = x * x + y * y;
  }

  // Current transform (uniform per batch)
  const float r00 = T[b * 6 + 0], r01 = T[b * 6 + 1];
  const float r10 = T[b * 6 + 2], r11 = T[b * 6 + 3];
  const float ttx = T[b * 6 + 4], tty = T[b * 6 + 5];

  const int rowBase = blockIdx.x * ROWS_PER_BLOCK + wave * 16;
  const int myRow   = rowBase + l15;
  const float sx  = src[((size_t)b * NPTS + myRow) * 2 + 0];
  const float sy  = src[((size_t)b * NPTS + myRow) * 2 + 1];
  const float stx = r00 * sx + r01 * sy + ttx;
  const float sty = r10 * sx + r11 * sy + tty;

  // A fragment (16x4 f32): lanes 0-15 -> K0,K1 = 2*st ; lanes 16-31 -> K2,K3 = 1,0
  v2f afrag;
  afrag.x = hi ? 1.0f : 2.0f * stx;
  afrag.y = hi ? 0.0f : 2.0f * sty;

  __syncthreads();

  // Online-softmax accumulators: 8 D-rows per lane (M=r and M=r+8 per half)
  float mx[8], ss[8], wx[8], wy[8];
#pragma unroll
  for (int r = 0; r < 8; ++r) { mx[r] = -1e30f; ss[r] = 0.f; wx[r] = 0.f; wy[r] = 0.f; }

  for (int m0 = 0; m0 < MPTS; m0 += 16) {
    const int col = m0 + l15;
    const float ctx = s_tx[col];
    const float cty = s_ty[col];
    // B fragment (4x16 f32): VGPR0 = rows K0 (lo lanes) / K2 (hi lanes)
    //                        VGPR1 = rows K1 (lo lanes) / K3 (hi lanes)
    v2f bfrag;
    bfrag.x = hi ? -s_tsq[col] : ctx;
    bfrag.y = hi ? 0.0f : cty;

    v8f acc = {};
    acc = __builtin_amdgcn_wmma_f32_16x16x4_f32(
        false, afrag, false, bfrag, (short)0, acc, false, false);

#pragma unroll
    for (int r = 0; r < 8; ++r) {
      float logit = acc[r];
      float nm   = fmaxf(mx[r], logit);
      float corr = __expf(mx[r] - nm);
      float e    = __expf(logit - nm);
      ss[r] = ss[r] * corr + e;
      wx[r] = wx[r] * corr + e * ctx;
      wy[r] = wy[r] * corr + e * cty;
      mx[r] = nm;
    }
  }

  // Merge lane-partial softmax accumulators within each 16-lane half
  // (fixed order -> bit-deterministic)
#pragma unroll
  for (int r = 0; r < 8; ++r) {
#pragma unroll
    for (int mask = 1; mask < 16; mask <<= 1) {
      float omx = __shfl_xor(mx[r], mask, 32);
      float oss = __shfl_xor(ss[r], mask, 32);
      float owx = __shfl_xor(wx[r], mask, 32);
      float owy = __shfl_xor(wy[r], mask, 32);
      float nm = fmaxf(mx[r], omx);
      float c1 = __expf(mx[r] - nm);
      float c2 = __expf(omx - nm);
      ss[r] = ss[r] * c1 + oss * c2;
      wx[r] = wx[r] * c1 + owx * c2;
      wy[r] = wy[r] * c1 + owy * c2;
      mx[r] = nm;
    }
  }

  // Publish st rows (lanes 0-15) and tc rows (lane r of each half -> row r / r+8)
  if (!hi) { s_stx[wave * 16 + l15] = stx; s_sty[wave * 16 + l15] = sty; }
#pragma unroll
  for (int r = 0; r < 8; ++r) {
    if (l15 == r) {
      int rr = wave * 16 + r + (hi ? 8 : 0);
      float inv = 1.0f / ss[r];
      s_tcx[rr] = wx[r] * inv;
      s_tcy[rr] = wy[r] * inv;
    }
  }
  __syncthreads();

  // Deterministic block reduction of {sum st, sum tc, sum st (x) tc}
  float v[8];
  if (tid < ROWS_PER_BLOCK) {
    float a0 = s_stx[tid], a1 = s_sty[tid];
    float b0 = s_tcx[tid], b1 = s_tcy[tid];
    v[0] = a0; v[1] = a1; v[2] = b0; v[3] = b1;
    v[4] = a0 * b0; v[5] = a0 * b1; v[6] = a1 * b0; v[7] = a1 * b1;
  } else {
#pragma unroll
    for (int r = 0; r < 8; ++r) v[r] = 0.f;
  }
#pragma unroll
  for (int r = 0; r < 8; ++r) {
#pragma unroll
    for (int mask = 1; mask < 32; mask <<= 1) v[r] += __shfl_xor(v[r], mask, 32);
  }
  if (lane == 0) {
#pragma unroll
    for (int r = 0; r < 8; ++r) s_red[r][wave] = v[r];
  }
  __syncthreads();
  if (tid < 8) {
    float s = 0.f;
    for (int w = 0; w < WAVES; ++w) s += s_red[tid][w];
    partial[(((size_t)b * NBLK) + blockIdx.x) * 8 + tid] = s;
  }
}

// ---------------------------------------------------------------------------
// Per-batch Kabsch update (closed-form 2x2 "SVD": theta = atan2(H01-H10, H00+H11))
// ---------------------------------------------------------------------------
__global__ void icp_update_kernel(float* __restrict__ T,
                                  const float* __restrict__ partial) {
  int b = threadIdx.x;
  if (b >= BATCH) return;
  float S[8] = {0, 0, 0, 0, 0, 0, 0, 0};
  for (int blk = 0; blk < NBLK; ++blk)
#pragma unroll
    for (int r = 0; r < 8; ++r)
      S[r] += partial[(((size_t)b * NBLK) + blk) * 8 + r];

  const float Nf = (float)NPTS, invN = 1.0f / (float)NPTS;
  float csx = S[0] * invN, csy = S[1] * invN;
  float ctx = S[2] * invN, cty = S[3] * invN;
  float H00 = S[4] - Nf * csx * ctx;
  float H01 = S[5] - Nf * csx * cty;
  float H10 = S[6] - Nf * csy * ctx;
  float H11 = S[7] - Nf * csy * cty;
  float th = atan2f(H01 - H10, H00 + H11);
  float c = cosf(th), s = sinf(th);
  float tdx = ctx - (c * csx - s * csy);
  float tdy = cty - (s * csx + c * csy);

  float r00 = T[b * 6 + 0], r01 = T[b * 6 + 1];
  float r10 = T[b * 6 + 2], r11 = T[b * 6 + 3];
  float px = T[b * 6 + 4], py = T[b * 6 + 5];
  T[b * 6 + 0] = c * r00 - s * r10;
  T[b * 6 + 1] = c * r01 - s * r11;
  T[b * 6 + 2] = s * r00 + c * r10;
  T[b * 6 + 3] = s * r01 + c * r11;
  T[b * 6 + 4] = c * px - s * py + tdx;
  T[b * 6 + 5] = s * px + c * py + tdy;
}

__global__ void icp_init_kernel(const float* __restrict__ init, float* __restrict__ T) {
  int b = threadIdx.x;
  if (b >= BATCH) return;
  float th = init[b * 3 + 0], tx = init[b * 3 + 1], ty = init[b * 3 + 2];
  float c = cosf(th), s = sinf(th);
  T[b * 6 + 0] = c;  T[b * 6 + 1] = -s;
  T[b * 6 + 2] = s;  T[b * 6 + 3] = c;
  T[b * 6 + 4] = tx; T[b * 6 + 5] = ty;
}

__global__ void icp_final_kernel(const float* __restrict__ T, float* __restrict__ out) {
  int b = threadIdx.x;
  if (b >= BATCH) return;
  out[b * 3 + 0] = atan2f(T[b * 6 + 2], T[b * 6 + 0]);
  out[b * 3 + 1] = T[b * 6 + 4];
  out[b * 3 + 2] = T[b * 6 + 5];
}

extern "C" void kernel_launch(void* const* d_in, const int* in_sizes, int n_in,
                              void* d_out, int out_size, void* d_ws, size_t ws_size,
                              hipStream_t stream) {
  const float* src  = (const float*)d_in[0];   // [32, 2048, 2]
  const float* tgt  = (const float*)d_in[1];   // [32, 2048, 2]
  const float* init = (const float*)d_in[2];   // [32, 3]
  float* out = (float*)d_out;                  // [32, 3]

  float* T       = (float*)d_ws;               // 32 * 6 floats
  float* partial = T + BATCH * 6;              // 32 * 16 * 8 floats

  icp_init_kernel<<<1, BATCH, 0, stream>>>(init, T);
  for (int it = 0; it < ITERS; ++it) {
    icp_corr_kernel<<<dim3(NBLK, BATCH), 256, 0, stream>>>(src, tgt, T, partial);
    icp_update_kernel<<<1, BATCH, 0, stream>>>(T, partial);
  }
  icp_final_kernel<<<1, BATCH, 0, stream>>>(T, out);
}